// DEFORM_sim_38388417692377
// MI455X (gfx1250) — compile-verified
//
#include <hip/hip_runtime.h>

#define VN 13
#define EN 12
#define HID 32
#define FEATP 848   // 844 padded to multiple of 4

typedef __attribute__((ext_vector_type(2))) float v2f;
typedef __attribute__((ext_vector_type(8))) float v8f;

struct V3 { float x, y, z; };
__device__ __forceinline__ V3 mkv(float a, float b, float c) { V3 r; r.x=a; r.y=b; r.z=c; return r; }
__device__ __forceinline__ V3 operator+(V3 a, V3 b) { return mkv(a.x+b.x, a.y+b.y, a.z+b.z); }
__device__ __forceinline__ V3 operator-(V3 a, V3 b) { return mkv(a.x-b.x, a.y-b.y, a.z-b.z); }
__device__ __forceinline__ V3 operator*(V3 a, float s) { return mkv(a.x*s, a.y*s, a.z*s); }
__device__ __forceinline__ float dot3(V3 a, V3 b) { return a.x*b.x + a.y*b.y + a.z*b.z; }
__device__ __forceinline__ V3 cross3(V3 a, V3 b) {
  return mkv(a.y*b.z - a.z*b.y, a.z*b.x - a.x*b.z, a.x*b.y - a.y*b.x);
}

// ---------------------------------------------------------------------------
// Phase 1: discrete-elastic-rod physics, one thread per batch element.
// Masked einsums reduced to banded 3-term sums + suffix sums.
// ---------------------------------------------------------------------------
__global__ __launch_bounds__(256)
void DEFORM_phys_kernel(const float* __restrict__ vert, const float* __restrict__ velo,
                        const float* __restrict__ rl_g, const float* __restrict__ rrl_g,
                        const float* __restrict__ rwp_g, const float* __restrict__ rwn_g,
                        const float* __restrict__ bend_g, const float* __restrict__ mass_g,
                        const float* __restrict__ ir_g, const int* __restrict__ clamped_g,
                        const int* __restrict__ pbd_g, float* __restrict__ pred_out, int B)
{
  int b = blockIdx.x * blockDim.x + threadIdx.x;
  if (b >= B) return;
  const float* Pv = vert + b * (VN * 3);
  const float* Vl = velo + b * (VN * 3);

  V3 x[VN], edges[EN], kb[EN], vec[EN];
  float rl[EN], denomf[EN], s1[EN], s2[EN];

#pragma unroll
  for (int v = 0; v < VN; ++v) x[v] = mkv(Pv[v*3], Pv[v*3+1], Pv[v*3+2]);
#pragma unroll
  for (int k = 0; k < EN; ++k) { rl[k] = rl_g[k]; edges[k] = x[k+1] - x[k]; }

  denomf[0] = 1.f; kb[0] = mkv(0.f, 0.f, 0.f);
#pragma unroll
  for (int k = 1; k < EN; ++k) {
    float den = rl[k-1]*rl[k] + dot3(edges[k-1], edges[k]);
    denomf[k] = den;
    kb[k] = cross3(edges[k-1], edges[k]) * (2.f / den);
  }

  // bishop frame scan fused with per-edge curvature coefficients
  V3 u = mkv(0.f,0.f,0.f), m1p = mkv(0.f,0.f,0.f), m2p = mkv(0.f,0.f,0.f);
  float dEdth = 0.f;
#pragma unroll
  for (int k = 0; k < EN; ++k) {
    float el = sqrtf(dot3(edges[k], edges[k])) + 1e-9f;
    V3 t = edges[k] * (1.f / el);
    if (k == 0) u = mkv(-t.y*t.x, 1.f - t.y*t.y, -t.y*t.z);
    else        u = u - t * dot3(u, t);
    u = u * (1.f / (sqrtf(dot3(u, u)) + 1e-9f));
    V3 m1c = u, m2c = cross3(t, u);
    float wpx =  dot3(kb[k], m2p), wpy = -dot3(kb[k], m1p);
    float wnx =  dot3(kb[k], m2c), wny = -dot3(kb[k], m1c);
    float rr  = rrl_g[k];
    float bp  = fmaxf(bend_g[k == 0 ? 0 : k-1], 1e-3f);
    float bc  = fmaxf(bend_g[k], 1e-3f);
    float d1x = bp * (wpx - rwp_g[2*k  ]) / rr;
    float d1y = bp * (wpy - rwp_g[2*k+1]) / rr;
    float d2x = bc * (wnx - rwn_g[2*k  ]) / rr;
    float d2y = bc * (wny - rwn_g[2*k+1]) / rr;
    vec[k] = m2p*d1x - m1p*d1y + m2c*d2x - m1c*d2y;    // sum of M^T d for both terms
    s1[k] = -d1x*wpy + d1y*wpx;                        // d1 . (J wp)
    s2[k] = -d2x*wny + d2y*wnx;                        // d2 . (J wn)
    if (k == EN-1) dEdth = s2[k];                      // twist term == s2 at last edge
    m1p = m1c; m2p = m2c;
  }

  float S1[EN+2], S2[EN+2];
  S1[EN] = S1[EN+1] = 0.f; S2[EN] = S2[EN+1] = 0.f;
#pragma unroll
  for (int k = EN-1; k >= 0; --k) { S1[k] = S1[k+1] + s1[k]; S2[k] = S2[k+1] + s2[k]; }

  float mass[VN], freev[VN];
#pragma unroll
  for (int v = 0; v < VN; ++v) { mass[v] = mass_g[v]; freev[v] = 1.f - (float)clamped_g[v]; }
  float ir = ir_g[0];

  V3 p[VN];
#pragma unroll
  for (int v = 0; v < VN; ++v) {
    V3 f = mkv(0.f, 0.f, 0.f);
    if (v >= 1 && v <= VN-2) {
      // GKB band: vec^T * {plusGKB[v-1], eqGKB[v], minusGKB[v+1]}, negated
      if (v >= 2) {
        int k = v-1; V3 a = vec[k]; V3 ep = edges[k-1];
        V3 g = (cross3(a, ep)*2.f - ep*dot3(a, kb[k])) * (1.f/denomf[k]);
        f = f - g;
      }
      {
        int k = v; V3 a = vec[k];
        V3 gm = (cross3(a, edges[k  ])*2.f + edges[k  ]*dot3(a, kb[k])) * (1.f/denomf[k]);
        V3 gp = (cross3(a, edges[k-1])*2.f - edges[k-1]*dot3(a, kb[k])) * (1.f/denomf[k]);
        f = f + gm + gp;          // -(vec^T eqGKB)
      }
      if (v <= VN-3) {
        int k = v+1; V3 a = vec[k];
        V3 g = (cross3(a, edges[k])*2.f + edges[k]*dot3(a, kb[k])) * (1.f/denomf[k]);
        f = f - g;
      }
      // holonomy gradients via suffix sums
      V3 ae = (kb[v]*(0.5f/rl[v-1]) + kb[v]*(-0.5f/rl[v])) * (-1.f);  // eqGH[v]
      V3 ghn = ae;
      f = f + ae*(S1[v+1] + S2[v]);
      if (v >= 2) {
        V3 ap = kb[v-1] * (0.5f / rl[v-2]);                           // plusGH[v-1]
        f = f + ap*(S1[v] + S2[v-1]);
        ghn = ghn + ap;
      }
      if (v <= VN-3) {
        V3 am = kb[v+1] * (-0.5f / rl[v+1]);                          // minusGH[v+1]
        f = f + am*(S1[v+2] + S2[v+1]);
        ghn = ghn + am;
      }
      f = f + ghn * dEdth;
      float fn = sqrtf(dot3(f, f));
      if (fn >= 5.0f) f = f * (5.0f / (fn + 1e-9f));
    }
    f = f * freev[v];
    V3 acc = f * (1.f/mass[v]) + mkv(0.f, 0.f, -9.81f);
    V3 vel = mkv(Vl[v*3], Vl[v*3+1], Vl[v*3+2]);
    vel = (vel + acc*(0.01f*ir)) * freev[v];
    p[v] = x[v] + vel*0.01f;
  }

  // PBD (Jacobi, mass-weighted)
  float winv[VN], wsum[EN];
#pragma unroll
  for (int v = 0; v < VN; ++v) winv[v] = freev[v] / mass[v];
#pragma unroll
  for (int k = 0; k < EN; ++k) wsum[k] = winv[k] + winv[k+1] + 1e-9f;
  int iters = pbd_g[0];
  for (int it = 0; it < iters; ++it) {
    V3 s[EN];
#pragma unroll
    for (int k = 0; k < EN; ++k) {
      V3 d = p[k+1] - p[k];
      float ln = sqrtf(dot3(d, d)) + 1e-9f;
      s[k] = d * ((ln - rl[k]) / (ln * wsum[k]));
    }
#pragma unroll
    for (int v = 0; v < VN; ++v) {
      V3 dd = mkv(0.f, 0.f, 0.f);
      if (v < VN-1) dd = dd + s[v]   * winv[v];
      if (v > 0)    dd = dd - s[v-1] * winv[v];
      p[v] = p[v] + dd;
    }
  }

  float* O = pred_out + b * (VN * 3);
#pragma unroll
  for (int v = 0; v < VN; ++v) { O[v*3] = p[v].x; O[v*3+1] = p[v].y; O[v*3+2] = p[v].z; }
}

// ---------------------------------------------------------------------------
// Phase 2: GCN + FC corrector. One wave per 16-batch tile; all channel-mixing
// matmuls via V_WMMA_F32_16X16X4_F32 (exact fp32); AHAT (tridiagonal chain
// normalization) applied with 3-term VALU stencil.
// ---------------------------------------------------------------------------
__device__ __forceinline__ v8f wmma4(v2f a, v2f b, v8f c) {
  return __builtin_amdgcn_wmma_f32_16x16x4_f32(false, a, false, b, (short)0, c, false, false);
}
__device__ __forceinline__ float ahat_d(int v) { return (v == 0 || v == VN-1) ? 0.5f : (1.f/3.f); }
__device__ __forceinline__ float ahat_o(int e) { return (e == 0 || e == VN-2) ? 0.4082482904638631f : (1.f/3.f); }

__global__ __launch_bounds__(32)
void DEFORM_gcnfc_kernel(const float* __restrict__ pred, const float* __restrict__ vert,
                         const float* __restrict__ W1v, const float* __restrict__ b1v,
                         const float* __restrict__ W2v, const float* __restrict__ b2v,
                         const float* __restrict__ W1d, const float* __restrict__ b1d,
                         const float* __restrict__ W2d, const float* __restrict__ b2d,
                         const float* __restrict__ fcW1, const float* __restrict__ fcb1,
                         const float* __restrict__ fcW2, const float* __restrict__ fcb2,
                         float* __restrict__ out)
{
  extern __shared__ float sm[];
  float* sP    = sm;                 // 16*39   = 624
  float* sIn   = sP    + 624;        // 208*4   = 832
  float* sMid  = sIn   + 832;        // 208*32  = 6656
  float* sFeat = sMid  + 6656;       // 16*848  = 13568
  float* sH1   = sFeat + 13568;      // 16*96   = 1536
  float* sRes  = sH1   + 1536;       // 16*27   = 432

  const int lane  = threadIdx.x;
  const int mrow  = lane & 15;       // A-row / B-col within tile
  const int khalf = lane >> 4;       // K-pair select (K=2*khalf,2*khalf+1)
  const int col   = lane & 15;       // C col
  const int rsh   = (lane >> 4) * 8; // C row base (rows j+rsh)
  const int b0    = blockIdx.x * 16;
  const float kmask = 1.f - (float)khalf;   // 1 for lanes 0-15, 0 for 16-31

  for (int i = lane; i < 624; i += 32) sP[i] = pred[b0*39 + i];

  for (int pass = 0; pass < 2; ++pass) {
    const float* W1 = pass ? W1d : W1v;
    const float* B1 = pass ? b1d : b1v;
    const float* W2 = pass ? W2d : W2v;
    const float* B2 = pass ? b2d : b2v;
    const int off = pass * (VN * HID);

    // stage 208x4 input rows (K padded with zero column)
    for (int i = lane; i < 16*VN; i += 32) {
      int bb = i / VN, v = i % VN;
      float vx = sP[bb*39 + v*3 + 0], vy = sP[bb*39 + v*3 + 1], vz = sP[bb*39 + v*3 + 2];
      if (pass) {
        vx -= vert[(b0+bb)*39 + v*3 + 0];
        vy -= vert[(b0+bb)*39 + v*3 + 1];
        vz -= vert[(b0+bb)*39 + v*3 + 2];
      }
      sIn[i*4+0] = vx; sIn[i*4+1] = vy; sIn[i*4+2] = vz; sIn[i*4+3] = 0.f;
    }

    // layer 1: T = X * W1  (208x3 @ 3x32, K padded to 4).
    // B fragments are loop-invariant: hoist; pad row (K=3) generated by an
    // always-valid load times a lane mask (no exec-predicated loads).
    v2f l1bA, l1bB;
    {
      int r0 = (2*khalf) * HID;                    // row 0 or row 2
      int r1 = (khalf ? 2 : 1) * HID;              // row 1, or row 2 masked to 0
      l1bA.x = W1[r0 + mrow];        l1bA.y = W1[r1 + mrow]      * kmask;
      l1bB.x = W1[r0 + 16 + mrow];   l1bB.y = W1[r1 + 16 + mrow] * kmask;
    }
    for (int mt = 0; mt < 13; ++mt) {
      v2f a;
      a.x = sIn[(mt*16 + mrow)*4 + 2*khalf + 0];
      a.y = sIn[(mt*16 + mrow)*4 + 2*khalf + 1];
      v8f acc0 = {0,0,0,0,0,0,0,0}, acc1 = {0,0,0,0,0,0,0,0};
      acc0 = wmma4(a, l1bA, acc0);
      acc1 = wmma4(a, l1bB, acc1);
      int rb = mt*16 + rsh;
#pragma unroll
      for (int j = 0; j < 8; ++j) {
        sMid[(rb+j)*HID + col]      = acc0[j];
        sMid[(rb+j)*HID + 16 + col] = acc1[j];
      }
    }

    // AHAT tridiagonal + bias + relu (in place, per (batch,channel) column)
    for (int pidx = lane; pidx < 16*HID; pidx += 32) {
      int bb = pidx >> 5, c = pidx & 31;
      float xv[VN];
#pragma unroll
      for (int v = 0; v < VN; ++v) xv[v] = sMid[(bb*VN + v)*HID + c];
#pragma unroll
      for (int v = 0; v < VN; ++v) {
        float y = ahat_d(v) * xv[v];
        if (v > 0)    y += ahat_o(v-1) * xv[v-1];
        if (v < VN-1) y += ahat_o(v)   * xv[v+1];
        y += B1[c];
        sMid[(bb*VN + v)*HID + c] = y > 0.f ? y : 0.f;
      }
    }

    // layer 2: (208x32)@(32x32), in place per M-tile
    for (int mt = 0; mt < 13; ++mt) {
      v8f acc0 = {0,0,0,0,0,0,0,0}, acc1 = {0,0,0,0,0,0,0,0};
#pragma unroll
      for (int k0 = 0; k0 < HID; k0 += 4) {
        int kr = k0 + 2*khalf;
        v2f a;
        a.x = sMid[(mt*16 + mrow)*HID + kr];
        a.y = sMid[(mt*16 + mrow)*HID + kr + 1];
        v2f bA, bB;
        bA.x = W2[kr*HID + mrow];      bA.y = W2[(kr+1)*HID + mrow];
        acc0 = wmma4(a, bA, acc0);
        bB.x = W2[kr*HID + 16 + mrow]; bB.y = W2[(kr+1)*HID + 16 + mrow];
        acc1 = wmma4(a, bB, acc1);
      }
      int rb = mt*16 + rsh;
#pragma unroll
      for (int j = 0; j < 8; ++j) {
        sMid[(rb+j)*HID + col]      = acc0[j];
        sMid[(rb+j)*HID + 16 + col] = acc1[j];
      }
    }

    // AHAT + bias + relu -> sFeat segment
    for (int pidx = lane; pidx < 16*HID; pidx += 32) {
      int bb = pidx >> 5, c = pidx & 31;
      float xv[VN];
#pragma unroll
      for (int v = 0; v < VN; ++v) xv[v] = sMid[(bb*VN + v)*HID + c];
#pragma unroll
      for (int v = 0; v < VN; ++v) {
        float y = ahat_d(v) * xv[v];
        if (v > 0)    y += ahat_o(v-1) * xv[v-1];
        if (v < VN-1) y += ahat_o(v)   * xv[v+1];
        y += B2[c];
        sFeat[bb*FEATP + off + v*HID + c] = y > 0.f ? y : 0.f;
      }
    }
  }

  // clamp-vertex features (cols 832..843) + zero pad (844..847)
  for (int i = lane; i < 16*16; i += 32) {
    int bb = i >> 4, j = i & 15;
    float val = 0.f;
    if (j < 12) {
      int vi = (j < 3) ? 0 : (j < 6) ? 1 : (j < 9) ? (VN-2) : (VN-1);
      val = sP[bb*39 + vi*3 + (j % 3)];
    }
    sFeat[bb*FEATP + 832 + j] = val;
  }

  // FC1: (16x848) @ (844x96) -> relu -> sH1
  {
    v8f acc[6];
#pragma unroll
    for (int nt = 0; nt < 6; ++nt) acc[nt] = (v8f){0,0,0,0,0,0,0,0};
#pragma unroll 1
    for (int k0 = 0; k0 < FEATP; k0 += 4) {
      int kr = k0 + 2*khalf;
      v2f a;
      a.x = sFeat[mrow*FEATP + kr];
      a.y = sFeat[mrow*FEATP + kr + 1];
      int ka = kr     < 843 ? kr     : 843;   // padded rows hit zero A cols; clamp B addr
      int kc = (kr+1) < 843 ? (kr+1) : 843;
      // pull the next K-block of fcW1 toward the caches while WMMAs issue
      int kp = (k0 + 64 < 844) ? (k0 + 64) : 843;
      __builtin_prefetch(&fcW1[kp*96 + mrow], 0, 1);
#pragma unroll
      for (int nt = 0; nt < 6; ++nt) {
        v2f bb; bb.x = fcW1[ka*96 + nt*16 + mrow]; bb.y = fcW1[kc*96 + nt*16 + mrow];
        acc[nt] = wmma4(a, bb, acc[nt]);
      }
    }
#pragma unroll
    for (int nt = 0; nt < 6; ++nt)
#pragma unroll
      for (int j = 0; j < 8; ++j) {
        float h = acc[nt][j] + fcb1[nt*16 + col];
        sH1[(rsh + j)*96 + nt*16 + col] = h > 0.f ? h : 0.f;
      }
  }

  // FC2: (16x96) @ (96x27) + fcb2 -> sRes
  {
    v8f r0 = {0,0,0,0,0,0,0,0}, r1 = {0,0,0,0,0,0,0,0};
    int c2  = 16 + mrow;
    int c2c = c2 < 27 ? c2 : 26;
    float mk = c2 < 27 ? 1.f : 0.f;
#pragma unroll 1
    for (int k0 = 0; k0 < 96; k0 += 4) {
      int kr = k0 + 2*khalf;
      v2f a;
      a.x = sH1[mrow*96 + kr];
      a.y = sH1[mrow*96 + kr + 1];
      v2f bA, bB;
      bA.x = fcW2[kr*27 + mrow];        bA.y = fcW2[(kr+1)*27 + mrow];
      r0 = wmma4(a, bA, r0);
      bB.x = fcW2[kr*27 + c2c] * mk;    bB.y = fcW2[(kr+1)*27 + c2c] * mk;
      r1 = wmma4(a, bB, r1);
    }
#pragma unroll
    for (int j = 0; j < 8; ++j) {
      int row = rsh + j;
      sRes[row*27 + col] = r0[j] + fcb2[col];
      if (16 + col < 27) sRes[row*27 + 16 + col] = r1[j] + fcb2[16 + col];
    }
  }

  // out = pred, with residual added at vertices 2..10
  for (int i = lane; i < 624; i += 32) {
    int bb = i / 39, r = i % 39, v = r / 3, c = r % 3;
    float val = sP[i];
    if (v >= 2 && v <= VN-3) val += sRes[bb*27 + (v-2)*3 + c];
    out[b0*39 + i] = val;
  }
}

// ---------------------------------------------------------------------------
extern "C" void kernel_launch(void* const* d_in, const int* in_sizes, int n_in,
                              void* d_out, int out_size, void* d_ws, size_t ws_size,
                              hipStream_t stream) {
  (void)n_in; (void)out_size; (void)ws_size;
  const float* vert  = (const float*)d_in[0];
  const float* velo  = (const float*)d_in[1];
  const float* rl    = (const float*)d_in[2];
  const float* rrl   = (const float*)d_in[3];
  const float* rwp   = (const float*)d_in[4];
  const float* rwn   = (const float*)d_in[5];
  const float* bend  = (const float*)d_in[6];
  const float* mass  = (const float*)d_in[7];
  const float* ir    = (const float*)d_in[8];
  const float* W1v   = (const float*)d_in[9];
  const float* b1v   = (const float*)d_in[10];
  const float* W2v   = (const float*)d_in[11];
  const float* b2v   = (const float*)d_in[12];
  const float* W1d   = (const float*)d_in[13];
  const float* b1d   = (const float*)d_in[14];
  const float* W2d   = (const float*)d_in[15];
  const float* b2d   = (const float*)d_in[16];
  const float* fcW1  = (const float*)d_in[17];
  const float* fcb1  = (const float*)d_in[18];
  const float* fcW2  = (const float*)d_in[19];
  const float* fcb2  = (const float*)d_in[20];
  const int* clamped = (const int*)d_in[21];
  const int* pbd     = (const int*)d_in[22];

  int B = in_sizes[0] / (VN * 3);
  float* pred = (float*)d_ws;

  DEFORM_phys_kernel<<<(B + 255) / 256, 256, 0, stream>>>(
      vert, velo, rl, rrl, rwp, rwn, bend, mass, ir, clamped, pbd, pred, B);

  int tiles = B / 16;
  size_t lds = (size_t)(624 + 832 + 6656 + 13568 + 1536 + 432) * sizeof(float);
  DEFORM_gcnfc_kernel<<<tiles, 32, lds, stream>>>(
      pred, vert, W1v, b1v, W2v, b2v, W1d, b1d, W2d, b2d,
      fcW1, fcb1, fcW2, fcb2, (float*)d_out);
}